// GATv2Attention_90039694393813
// MI455X (gfx1250) — compile-verified
//
#include <hip/hip_runtime.h>
#include <stdint.h>

// Problem constants (reference: B=4,H=8,LQ=256,LK=256,D=64)
#define B_   4
#define H_   8
#define LQ_  256
#define LK_  256
#define D_   64
#define ITILE 16           // query rows per workgroup
#define KROW  65           // padded LDS row stride (floats) for K tile -> conflict-free
#define SROW  257          // padded LDS row stride (floats) for score tile

typedef float    v2f __attribute__((ext_vector_type(2)));
typedef float    v8f __attribute__((ext_vector_type(8)));
typedef uint32_t v4u __attribute__((ext_vector_type(4)));
typedef int      v4i __attribute__((ext_vector_type(4)));
typedef int      v8i __attribute__((ext_vector_type(8)));

#if defined(__gfx1250__) && __has_builtin(__builtin_amdgcn_tensor_load_to_lds)
#define USE_TDM 1
#else
#define USE_TDM 0
#endif

#if defined(__gfx1250__) && __has_builtin(__builtin_amdgcn_wmma_f32_16x16x4_f32)
#define USE_WMMA 1
#else
#define USE_WMMA 0
#endif

__device__ __forceinline__ float tanh_fast(float x) {
#if defined(__gfx1250__) && __has_builtin(__builtin_amdgcn_tanhf)
    return __builtin_amdgcn_tanhf(x);          // v_tanh_f32: 1 transcendental
#else
    // tanh(x) = 1 - 2/(e^{2x}+1) ; e^{2x} = 2^{2x*log2 e}
    float e = __builtin_amdgcn_exp2f(x * 2.8853900817779268f);
    return 1.0f - 2.0f * __builtin_amdgcn_rcpf(e + 1.0f);
#endif
}

__global__ __launch_bounds__(256)
void gatv2_attn_kernel(const float* __restrict__ q,
                       const float* __restrict__ k,
                       const unsigned char* __restrict__ mask,  // bool (1B), True = padded
                       const float* __restrict__ attn_w,        // (1,H,1,1,D)
                       float* __restrict__ out) {
    __shared__ float k_lds[LK_ * KROW];     // K tile, row stride 65 floats (TDM-padded)
    __shared__ float q2_s[ITILE * D_];      // 0.5*q tile
    __shared__ float a_s[D_];               // attention weights for this head
    __shared__ float s_lds[ITILE * SROW];   // scores, then exp values (in place)
    __shared__ float pmax[ITILE * 16];
    __shared__ float psum[ITILE * 16];
    __shared__ float rowmax[ITILE];
    __shared__ float rowsum[ITILE];

    const int t    = threadIdx.x;
    const int tile = blockIdx.x & 15;       // LQ/ITILE = 16 tiles
    const int bh   = blockIdx.x >> 4;       // b*H + h
    const int b    = bh >> 3;               // H = 8
    const int h    = bh & 7;
    const int i0   = tile * ITILE;

    const float*         kt = k + (size_t)bh * (LK_ * D_);
    const float*         qt = q + (size_t)bh * (LQ_ * D_) + (size_t)i0 * D_;
    const unsigned char* mt = mask + (size_t)b * (LQ_ * LK_) + (size_t)i0 * LK_;

    __builtin_prefetch(mt, 0, 0);           // global_prefetch_b8
    __builtin_prefetch(qt, 0, 0);

    // ---- Stage q/2 and a into LDS ----
    for (int e = t; e < ITILE * D_; e += 256) q2_s[e] = 0.5f * qt[e];
    if (t < D_) a_s[t] = attn_w[h * D_ + t];

    // ---- Stage K tile (256x64 f32) into padded LDS via Tensor Data Mover ----
#if USE_TDM
    if (t == 0) {
        uint64_t ga = (uint64_t)(uintptr_t)kt;
        uint32_t lo = (uint32_t)(uintptr_t)&k_lds[0];   // flat-LDS low 32 bits = LDS offset
        // D# group0: count=1, lds_addr, global_addr[56:0], type=2 (bits 127:126)
        v4u g0 = { 1u, lo, (uint32_t)ga,
                   (uint32_t)((ga >> 32) & 0x01FFFFFFu) | (2u << 30) };
        // D# group1:
        //  d0: data_size=2 (4B) | pad_enable | pad_interval=5 (64 DW) | pad_amount=0 (1 DW)
        //  tensor_dim0=64, tensor_dim1=256, tile_dim0=64, tile_dim1=256, tile_dim2=0
        //  dim0_stride=64, dim1_stride=16384 (unused for 2D)
        v8i g1 = { (int)0x01520000,
                   (int)(64u  << 16),   // tensor_dim0[15:0] in bits 63:48
                   (int)(256u << 16),   // tensor_dim1[15:0] in bits 95:80
                   (int)(64u  << 16),   // tile_dim0 in bits 127:112
                   256,                 // tile_dim1
                   64,                  // tensor_dim0_stride[31:0]
                   (int)(16384u << 16), // tensor_dim1_stride[15:0]
                   0 };
        v4i z4 = { 0, 0, 0, 0 };
#if __has_include(<hip/amd_detail/amd_gfx1250_TDM.h>)
        v8i z8 = { 0, 0, 0, 0, 0, 0, 0, 0 };
        __builtin_amdgcn_tensor_load_to_lds(g0, g1, z4, z4, z8, 0);  // 6-arg toolchain
#else
        __builtin_amdgcn_tensor_load_to_lds(g0, g1, z4, z4, 0);      // 5-arg ROCm 7.2
#endif
        __builtin_amdgcn_s_wait_tensorcnt(0);
    }
#else
    for (int e = t; e < LK_ * D_; e += 256)
        k_lds[(e >> 6) * KROW + (e & 63)] = kt[e];
#endif
    __syncthreads();

    // ---- Main score loop: lane owns key j = t; 16 query rows accumulated ----
    // silu(x)*a = a*xh*(1+tanh(xh)) with xh = x/2 = q/2 + k/2
    float acc[ITILE];
#pragma unroll
    for (int i = 0; i < ITILE; ++i) acc[i] = 0.0f;

    const int j = t;   // 256 threads == 256 keys
#pragma unroll 2
    for (int d = 0; d < D_; ++d) {
        float kh = 0.5f * k_lds[j * KROW + d];   // per-lane, bank-conflict-free
        float ad = a_s[d];                        // broadcast
#pragma unroll
        for (int i = 0; i < ITILE; ++i) {
            float xh = q2_s[i * D_ + d] + kh;     // broadcast + add
            float th = tanh_fast(xh);
            float v  = __builtin_fmaf(xh, th, xh);        // xh*(1+tanh)
            acc[i]   = __builtin_fmaf(ad, v, acc[i]);
        }
    }

    // ---- Apply mask, write score tile (16 x 256) to LDS ----
#pragma unroll
    for (int i = 0; i < ITILE; ++i) {
        bool m = mt[i * LK_ + j] != 0;
        s_lds[i * SROW + j] = m ? -3.402823466e38f : acc[i];
    }
    __syncthreads();

    // ---- Row max (two-stage LDS reduction) ----
    const int row = t >> 4, g = t & 15;
    const int base = row * SROW + g * 16;
    {
        float lm = -3.402823466e38f;
#pragma unroll
        for (int c = 0; c < 16; ++c) lm = fmaxf(lm, s_lds[base + c]);
        pmax[row * 16 + g] = lm;
    }
    __syncthreads();
    if (t < ITILE) {
        float m = pmax[t * 16];
#pragma unroll
        for (int c = 1; c < 16; ++c) m = fmaxf(m, pmax[t * 16 + c]);
        rowmax[t] = m;
    }
    __syncthreads();

    // ---- exp(x - max) in place + partial sums ----
    {
        float rm = rowmax[row];
        float ls = 0.0f;
#pragma unroll
        for (int c = 0; c < 16; ++c) {
            float e = __builtin_amdgcn_exp2f((s_lds[base + c] - rm) * 1.4426950408889634f);
            s_lds[base + c] = e;
            ls += e;
        }
        psum[row * 16 + g] = ls;
    }
    __syncthreads();

    // ---- Row sums of the 16x256 exp tile ----
#if USE_WMMA
    // One wave: C(16x16) += A(16x4) x B(4x16=ones), 64 K-chunks -> every column = row sum.
    if (t < 32) {
        const int r16 = t & 15;
        const int hi  = t >> 4;
        const int co  = hi ? 2 : 0;        // lanes 16-31 hold K=2,3 of the A chunk
        v2f bv = { 1.0f, 1.0f };           // all-ones B (layout-agnostic)
        v8f cacc = {};
        for (int c = 0; c < 64; ++c) {
            v2f av;
            av.x = s_lds[r16 * SROW + 4 * c + co];
            av.y = s_lds[r16 * SROW + 4 * c + co + 1];
            cacc = __builtin_amdgcn_wmma_f32_16x16x4_f32(
                false, av, false, bv, (short)0, cacc, false, false);
        }
        if (r16 == 0) {                    // lane 0 -> M 0..7, lane 16 -> M 8..15
            int mb = hi ? 8 : 0;
#pragma unroll
            for (int r = 0; r < 8; ++r) rowsum[mb + r] = cacc[r];
        }
    }
#else
    if (t < ITILE) {
        float s = 0.0f;
#pragma unroll
        for (int c = 0; c < 16; ++c) s += psum[t * 16 + c];
        rowsum[t] = s;
    }
#endif
    __syncthreads();

    // ---- Normalize, re-zero masked entries, store (coalesced 64B segments) ----
    {
        float rinv = 1.0f / rowsum[row];
        float* orow = out + (size_t)bh * (LQ_ * LK_) + (size_t)(i0 + row) * LK_;
        const unsigned char* mrow = mt + row * LK_;
#pragma unroll
        for (int c = 0; c < 16; ++c) {
            int jj = g + 16 * c;
            float vv = s_lds[row * SROW + jj];
            orow[jj] = (mrow[jj] != 0) ? 0.0f : vv * rinv;
        }
    }
}

extern "C" void kernel_launch(void* const* d_in, const int* in_sizes, int n_in,
                              void* d_out, int out_size, void* d_ws, size_t ws_size,
                              hipStream_t stream) {
    // setup_inputs order: q(f32), k(f32), scale(f32, unused), mask(bool), attention(f32)
    const float*         q    = (const float*)d_in[0];
    const float*         k    = (const float*)d_in[1];
    const unsigned char* mask = (const unsigned char*)d_in[3];
    const float*         aw   = (const float*)d_in[4];
    float*               out  = (float*)d_out;

    dim3 grid(B_ * H_ * (LQ_ / ITILE));   // 512 workgroups
    gatv2_attn_kernel<<<grid, 256, 0, stream>>>(q, k, mask, aw, out);
}